// Attention_29463475650793
// MI455X (gfx1250) — compile-verified
//
#include <hip/hip_runtime.h>
#include <hip/hip_bf16.h>

typedef __attribute__((ext_vector_type(2))) float v2f;
typedef __attribute__((ext_vector_type(8))) float v8f;

// exact pointee type expected by __builtin_amdgcn_global_load_async_to_lds_b128
// (per hipcc diagnostic: "__attribute__((__vector_size__(4 * sizeof(int)))) int __device__ *")
typedef int vs4i __attribute__((vector_size(16)));

constexpr int H2C = 1024;   // 2*hidden
constexpr int SL  = 1024;   // seq len
constexpr int BB  = 64;     // batch

// scores kernel tiling
constexpr int TM   = 64;    // tokens per workgroup (4 WMMA M tiles)
constexpr int TO   = 256;   // output features per workgroup (16 o-chunks, 2/wave)
constexpr int KC   = 64;    // K-chunk staged in LDS
constexpr int NKC  = H2C / KC;          // 16 chunks
constexpr int LDST = KC + 4;            // padded LDS row stride (272B -> no bank conflicts)
constexpr int NPER = (TM * KC) / (256 * 4);  // B128 async ops per thread per chunk = 4

#if __has_builtin(__builtin_amdgcn_global_load_async_to_lds_b128) && \
    __has_builtin(__builtin_amdgcn_s_wait_asynccnt)
#define USE_ASYNC_LDS 1
#else
#define USE_ASYNC_LDS 0
#endif

// ---------------------------------------------------------------------------
// dec_fea = s_t_hat @ W_s^T + b_s  [B,H2];  also zero-inits the scores buffer.
// grid (8,4): blockIdx.y = M-tile (16 batch rows), blockIdx.x*8+wave = o-chunk
// ---------------------------------------------------------------------------
__global__ __launch_bounds__(256)
void decfea_kernel(const float* __restrict__ sth,  // [B,H2]
                   const float* __restrict__ Ws,   // [H2,H2]
                   const float* __restrict__ bs,   // [H2]
                   float* __restrict__ dec,        // [B,H2]
                   float* __restrict__ scores)     // [B,S]  (zeroed here)
{
    const int tid   = threadIdx.x;
    const int wave  = tid >> 5;
    const int lane  = tid & 31;
    const int l16   = lane & 15;
    const int lhalf = lane >> 4;

    // zero the scores accumulator buffer (32 blocks * 256 thr = 8192 threads)
    const int gid = blockIdx.y * (int)gridDim.x * 256 + blockIdx.x * 256 + tid;
    for (int i = gid; i < BB * SL; i += 32 * 256) scores[i] = 0.0f;

    const int mtile = blockIdx.y;                 // 0..3  (16 batch rows each)
    const int o0    = (blockIdx.x * 8 + wave) * 16;
    const int bi    = mtile * 16 + l16;

    // A 16x4 f32 layout: lanes<16 -> K = lhalf*2 + {0,1}; lanes>=16 -> +2
    const float* aRow = sth + (size_t)bi * H2C + lhalf * 2;
    const float* bRow = Ws + (size_t)(o0 + l16) * H2C + lhalf * 2;  // B[k,n]=Ws[o0+n,k]

    v8f c = {};
    for (int k = 0; k < H2C; k += 16) {
#pragma unroll
        for (int kk = 0; kk < 16; kk += 4) {
            v2f a = *(const v2f*)(aRow + k + kk);
            v2f b = *(const v2f*)(bRow + k + kk);
            c = __builtin_amdgcn_wmma_f32_16x16x4_f32(false, a, false, b,
                                                      (short)0, c, false, false);
        }
    }

    const int o    = o0 + l16;
    const float bias = bs[o];
#pragma unroll
    for (int r = 0; r < 8; ++r) {
        int brow = mtile * 16 + r + 8 * lhalf;    // C layout: lanes>=16 hold M=r+8
        dec[(size_t)brow * H2C + o] = c[r] + bias;
    }
}

// ---------------------------------------------------------------------------
// partial scores: for 64 tokens x 256 o-features per WG,
//   scores[b,s] += sum_{o in tile} v[o] * tanh(enc@Wh^T + dec + cov*Wc)
// grid (SL/TM=16, H2/TO=4, B=64), 256 threads (8 waves).
// A (enc tile) double-buffered in LDS via async loads; B (Wh) streamed from L2.
// Each wave: 4 M-tiles x 2 o-chunks = 8 live f32 accumulators, B reused across M.
// ---------------------------------------------------------------------------
__global__ __launch_bounds__(256)
void scores_kernel(const float* __restrict__ enc,   // [B,S,H2]
                   const float* __restrict__ Wh,    // [H2,H2]
                   const float* __restrict__ dec,   // [B,H2]
                   const float* __restrict__ Wc,    // [H2]
                   const float* __restrict__ vvec,  // [H2]
                   const float* __restrict__ cov,   // [B,S]
                   float* __restrict__ scores)      // [B,S] (atomic accumulate)
{
    __shared__ float ldsA[2][TM * LDST];   // 2 x 17408 B, padded rows
    __shared__ float ldsScore[TM];
    __shared__ float ldsCov[TM];

    const int b     = blockIdx.z;
    const int s0    = blockIdx.x * TM;
    const int oBase = blockIdx.y * TO;
    const int tid   = threadIdx.x;
    const int wave  = tid >> 5;
    const int lane  = tid & 31;
    const int l16   = lane & 15;
    const int lhalf = lane >> 4;

    if (tid < TM) {
        ldsScore[tid] = 0.0f;
        ldsCov[tid]   = cov[(size_t)b * SL + s0 + tid];
    }

    const float* encBase = enc + ((size_t)b * SL + s0) * H2C;

    // ---- staging helpers (chunk = TM x KC of enc, padded rows in LDS) ----
    auto stage = [&](float* buf, int k0) {
#if USE_ASYNC_LDS
#pragma unroll
        for (int q = 0; q < NPER; ++q) {
            int i    = tid + 256 * q;          // 0..1023 float4 slots
            int row  = i >> 4;                 // / (KC/4)
            int col4 = i & 15;
            const float* g = encBase + (size_t)row * H2C + k0 + col4 * 4;
            float*       l = buf + row * LDST + col4 * 4;
            __builtin_amdgcn_global_load_async_to_lds_b128(
                (__attribute__((address_space(1))) vs4i*)
                    (__attribute__((address_space(1))) void*)g,
                (__attribute__((address_space(3))) vs4i*)
                    (__attribute__((address_space(3))) void*)l,
                0, 0);
        }
#else
#pragma unroll
        for (int q = 0; q < NPER; ++q) {
            int i    = tid + 256 * q;
            int row  = i >> 4;
            int col4 = i & 15;
            *(float4*)(buf + row * LDST + col4 * 4) =
                *(const float4*)(encBase + (size_t)row * H2C + k0 + col4 * 4);
        }
#endif
    };

    v8f acc[8];
#pragma unroll
    for (int t = 0; t < 8; ++t) { v8f z = {}; acc[t] = z; }

    const int oA = oBase + (wave * 2 + 0) * 16;
    const int oB = oBase + (wave * 2 + 1) * 16;

    stage(&ldsA[0][0], 0);                      // prologue: chunk 0 in flight

    for (int c = 0; c < NKC; ++c) {
        const int cur = c & 1;
        __syncthreads();                        // everyone done reading buf[cur^1]
        if (c + 1 < NKC) {
            stage(&ldsA[cur ^ 1][0], (c + 1) * KC);
#if USE_ASYNC_LDS
            __builtin_amdgcn_s_wait_asynccnt(NPER);   // chunk c landed (in-order)
#endif
        } else {
#if USE_ASYNC_LDS
            __builtin_amdgcn_s_wait_asynccnt(0);
#endif
        }
        __syncthreads();                        // chunk c visible to all waves

        const float* aPtr  = &ldsA[cur][0] + l16 * LDST + lhalf * 2;
        const int    k0    = c * KC;
        const float* bPtr0 = Wh + (size_t)(oA + l16) * H2C + k0 + lhalf * 2;
        const float* bPtr1 = Wh + (size_t)(oB + l16) * H2C + k0 + lhalf * 2;

        for (int ks = 0; ks < KC; ks += 4) {
            v2f b0 = *(const v2f*)(bPtr0 + ks);
            v2f b1 = *(const v2f*)(bPtr1 + ks);
#pragma unroll
            for (int mt = 0; mt < 4; ++mt) {
                v2f a = *(const v2f*)(aPtr + mt * 16 * LDST + ks);   // ds_load_b64
                acc[mt * 2 + 0] = __builtin_amdgcn_wmma_f32_16x16x4_f32(
                    false, a, false, b0, (short)0, acc[mt * 2 + 0], false, false);
                acc[mt * 2 + 1] = __builtin_amdgcn_wmma_f32_16x16x4_f32(
                    false, a, false, b1, (short)0, acc[mt * 2 + 1], false, false);
            }
        }
    }

    __syncthreads();
    // fused epilogue: tanh(+dec+cov*Wc) dotted with v, reduced over o into LDS
#pragma unroll
    for (int j = 0; j < 2; ++j) {
        const int o   = (j == 0 ? oA : oB) + l16;
        const float dv  = dec[(size_t)b * H2C + o];
        const float wcv = Wc[o];
        const float vv  = vvec[o];
#pragma unroll
        for (int mt = 0; mt < 4; ++mt) {
#pragma unroll
            for (int r = 0; r < 8; ++r) {
                int m   = mt * 16 + r + 8 * lhalf;       // token within tile
                float e = tanhf(acc[mt * 2 + j][r] + dv + ldsCov[m] * wcv);
                atomicAdd(&ldsScore[m], e * vv);         // ds_add_f32
            }
        }
    }
    __syncthreads();
    if (tid < TM)
        atomicAdd(&scores[(size_t)b * SL + s0 + tid], ldsScore[tid]);
}

// ---------------------------------------------------------------------------
// Masked softmax + coverage update; also zeroes c_t for the context kernel.
// softmax(s)*mask / sum == exp(s-max)*mask / sum(exp(s-max)*mask)
// ---------------------------------------------------------------------------
__global__ __launch_bounds__(256)
void softmax_kernel(const float* __restrict__ scores,  // [B,S]
                    const float* __restrict__ mask,    // [B,S]
                    const float* __restrict__ cov,     // [B,S]
                    float* __restrict__ out)           // c_t | attn | cov_new
{
    __shared__ float red[256];
    __shared__ float pbuf[SL];

    const int b   = blockIdx.x;
    const int tid = threadIdx.x;
    float* c_t  = out;
    float* attn = out + (size_t)BB * H2C;
    float* covn = attn + (size_t)BB * SL;

    for (int i = tid; i < H2C; i += 256) c_t[(size_t)b * H2C + i] = 0.0f;

    float mx = -1e30f;
    for (int i = tid; i < SL; i += 256)
        mx = fmaxf(mx, scores[(size_t)b * SL + i]);
    red[tid] = mx;
    __syncthreads();
    for (int s = 128; s > 0; s >>= 1) {
        if (tid < s) red[tid] = fmaxf(red[tid], red[tid + s]);
        __syncthreads();
    }
    mx = red[0];
    __syncthreads();

    float sum = 0.0f;
    for (int i = tid; i < SL; i += 256) {
        float p = __expf(scores[(size_t)b * SL + i] - mx) * mask[(size_t)b * SL + i];
        pbuf[i] = p;
        sum += p;
    }
    red[tid] = sum;
    __syncthreads();
    for (int s = 128; s > 0; s >>= 1) {
        if (tid < s) red[tid] += red[tid + s];
        __syncthreads();
    }
    const float inv = 1.0f / red[0];

    for (int i = tid; i < SL; i += 256) {
        float a = pbuf[i] * inv;
        attn[(size_t)b * SL + i] = a;
        covn[(size_t)b * SL + i] = cov[(size_t)b * SL + i] + a;
    }
}

// ---------------------------------------------------------------------------
// c_t[b,o] += sum_s attn[b,s] * enc[b,s,o]   (memory-bound pass #2 over enc)
// grid (SL/128, H2/128, B), 128 threads; attn chunk staged in LDS.
// ---------------------------------------------------------------------------
__global__ __launch_bounds__(128)
void context_kernel(const float* __restrict__ enc,   // [B,S,H2]
                    const float* __restrict__ attn,  // [B,S]
                    float* __restrict__ c_t)         // [B,H2]
{
    __shared__ float la[128];
    const int b  = blockIdx.z;
    const int s0 = blockIdx.x * 128;
    const int o  = blockIdx.y * 128 + threadIdx.x;

    la[threadIdx.x] = attn[(size_t)b * SL + s0 + threadIdx.x];
    __syncthreads();

    float accv = 0.0f;
    const float* ep = enc + ((size_t)b * SL + s0) * H2C + o;
#pragma unroll 4
    for (int s = 0; s < 128; ++s)
        accv += la[s] * ep[(size_t)s * H2C];

    atomicAdd(&c_t[(size_t)b * H2C + o], accv);   // global_atomic_add_f32
}

// ---------------------------------------------------------------------------
extern "C" void kernel_launch(void* const* d_in, const int* in_sizes, int n_in,
                              void* d_out, int out_size, void* d_ws, size_t ws_size,
                              hipStream_t stream) {
    const float* sth  = (const float*)d_in[0];  // s_t_hat          [B,H2]
    const float* enc  = (const float*)d_in[1];  // encoder_outputs  [B,S,H2]
    const float* mask = (const float*)d_in[2];  // enc_padding_mask [B,S]
    const float* cov  = (const float*)d_in[3];  // coverage         [B,S]
    const float* Wh   = (const float*)d_in[4];  // [H2,H2]
    const float* Ws   = (const float*)d_in[5];  // [H2,H2]
    const float* bs   = (const float*)d_in[6];  // [H2]
    const float* Wc   = (const float*)d_in[7];  // [H2,1]
    const float* v    = (const float*)d_in[8];  // [1,H2]

    float* out    = (float*)d_out;              // c_t | attn | coverage_new
    float* dec    = (float*)d_ws;               // [B,H2]
    float* scores = dec + (size_t)BB * H2C;     // [B,S]

    decfea_kernel<<<dim3(8, 4), 256, 0, stream>>>(sth, Ws, bs, dec, scores);
    scores_kernel<<<dim3(SL / TM, H2C / TO, BB), 256, 0, stream>>>(
        enc, Wh, dec, Wc, v, cov, scores);
    softmax_kernel<<<BB, 256, 0, stream>>>(scores, mask, cov, out);
    context_kernel<<<dim3(SL / 128, H2C / 128, BB), 128, 0, stream>>>(
        enc, out + (size_t)BB * H2C, out);
}